// _LBlock_36747740184854
// MI455X (gfx1250) — compile-verified
//
#include <hip/hip_runtime.h>
#include <math.h>

#define N_NODES 100000
#define DIM     128
#define NE      600000
#define NB      64
#define NH      4
#define DH      32
#define SLOPE   0.2f
#define BN_EPS  1e-5f
#define TILE_M  32

typedef float v2f __attribute__((ext_vector_type(2)));
typedef float v8f __attribute__((ext_vector_type(8)));

__device__ __forceinline__ float lrelu(float x) { return x > 0.f ? x : SLOPE * x; }
__device__ __forceinline__ float geluf(float x) { return 0.5f * x * (1.f + erff(x * 0.70710678118654752f)); }
__device__ __forceinline__ float mfin(float m)  { return (m == -INFINITY) ? 0.f : m; }

__device__ __forceinline__ void atomicMaxFloat(float* addr, float v) {
  if (v >= 0.f) atomicMax((int*)addr, __float_as_int(v));
  else          atomicMin((unsigned int*)addr, __float_as_uint(v));
}

// ---------------------------------------------------------------------------
// C[M x Nc] = A[M x K] @ W[K x Nc] (+bias). M%32==0, K/Nc compile-time.
// 256 threads = 8 waves. Block computes 32 rows x 128 cols: each wave owns two
// 16x16 output tiles stacked in M that share one B fragment per WMMA step
// (halves B traffic). A tile staged into LDS with CDNA5 async-to-LDS loads
// (ASYNCcnt), shared by all 8 waves. K loop fully unrolled so all global/LDS
// fragment loads use immediate offsets. grid = (M/32, Nc/128)
// ---------------------------------------------------------------------------
template<int K, int Nc>
__global__ __launch_bounds__(256) void gemm_wmma(
    const float* __restrict__ A, const float* __restrict__ W,
    const float* __restrict__ bias, float* __restrict__ C)
{
  __shared__ float As[TILE_M * K];
  const int row0 = blockIdx.x * TILE_M;
  const int wave = threadIdx.x >> 5;
  const int lane = threadIdx.x & 31;
  const int col0 = (Nc > 128 ? (int)blockIdx.y * 128 : 0) + wave * 16;

  // async cooperative stage of the 32xK A tile into LDS (b128 per lane)
  const float* Ab = A + (size_t)row0 * K;
  for (int i = threadIdx.x * 4; i < TILE_M * K; i += 1024) {
    unsigned lds = (unsigned)(size_t)&As[i];
    const float* g = Ab + i;
    asm volatile("global_load_async_to_lds_b128 %0, %1, off"
                 :: "v"(lds), "v"(g) : "memory");
  }
  asm volatile("s_wait_asynccnt 0x0" ::: "memory");
  __syncthreads();

  const int m  = lane & 15;             // A row within tile / B column
  const int kk = (lane >> 4) << 1;      // K sub-offset 0 or 2 (ISA frag layout)
  const float* Wc = W + (size_t)kk * Nc + col0 + m;
  const float* A0 = &As[m * K + kk];
  const float* A1 = &As[(m + 16) * K + kk];
  v8f acc0 = {0.f,0.f,0.f,0.f,0.f,0.f,0.f,0.f};
  v8f acc1 = {0.f,0.f,0.f,0.f,0.f,0.f,0.f,0.f};

  #pragma unroll
  for (int k0 = 0; k0 < K; k0 += 4) {
    v2f b;  b.x  = Wc[k0 * Nc];  b.y  = Wc[k0 * Nc + Nc];
    v2f a0; a0.x = A0[k0];       a0.y = A0[k0 + 1];
    v2f a1; a1.x = A1[k0];       a1.y = A1[k0 + 1];
    acc0 = __builtin_amdgcn_wmma_f32_16x16x4_f32(false, a0, false, b,
                                                 (short)0, acc0, false, false);
    acc1 = __builtin_amdgcn_wmma_f32_16x16x4_f32(false, a1, false, b,
                                                 (short)0, acc1, false, false);
  }

  const int ccol  = col0 + m;
  const float bv  = bias ? bias[ccol] : 0.f;
  const int r0    = row0 + ((lane >> 4) << 3);
  #pragma unroll
  for (int r = 0; r < 8; ++r) {
    C[(size_t)(r0 + r)      * Nc + ccol] = acc0[r] + bv;
    C[(size_t)(r0 + 16 + r) * Nc + ccol] = acc1[r] + bv;
  }
}

// ---------------------------------------------------------------------------
__global__ void fill_f32(float* p, float v, long n) {
  long i = (long)blockIdx.x * blockDim.x + threadIdx.x;
  if (i < n) p[i] = v;
}

__global__ void gelu_k(const float* __restrict__ in, float* __restrict__ out, long n) {
  long i = (long)blockIdx.x * blockDim.x + threadIdx.x;
  if (i < n) out[i] = geluf(in[i]);
}

// per (node, head): attention logit contributions
__global__ void gat_logits(const float* __restrict__ h,
                           const float* __restrict__ asrc,
                           const float* __restrict__ adst,
                           float* __restrict__ alsrc, float* __restrict__ aldst)
{
  int idx = blockIdx.x * blockDim.x + threadIdx.x;
  if (idx >= N_NODES * NH) return;
  int n = idx >> 2, hh = idx & 3;
  const float4* hp = (const float4*)(h + (size_t)n * DIM + hh * DH);
  const float4* as = (const float4*)(asrc + hh * DH);
  const float4* ad = (const float4*)(adst + hh * DH);
  float ss = 0.f, sd = 0.f;
  #pragma unroll
  for (int i = 0; i < DH / 4; ++i) {
    float4 hv = hp[i], av = as[i], dv = ad[i];
    ss += hv.x * av.x + hv.y * av.y + hv.z * av.z + hv.w * av.w;
    sd += hv.x * dv.x + hv.y * dv.y + hv.z * dv.z + hv.w * dv.w;
  }
  alsrc[idx] = ss; aldst[idx] = sd;
}

// edges e >= NE are implicit self-loops
__global__ void gat_edge_max(const int* __restrict__ src, const int* __restrict__ dst,
                             const float* __restrict__ alsrc, const float* __restrict__ aldst,
                             float* __restrict__ elog, float* __restrict__ mbuf, int Eg)
{
  int idx = blockIdx.x * blockDim.x + threadIdx.x;
  if (idx >= Eg * NH) return;
  int e = idx >> 2, hh = idx & 3;
  int s = (e < NE) ? src[e] : (e - NE);
  int d = (e < NE) ? dst[e] : (e - NE);
  float v = lrelu(alsrc[s * NH + hh] + aldst[d * NH + hh]);
  elog[idx] = v;
  atomicMaxFloat(&mbuf[d * NH + hh], v);
}

__global__ void gat_edge_exps(const int* __restrict__ dst,
                              float* __restrict__ elog, const float* __restrict__ mbuf,
                              float* __restrict__ sbuf, int Eg)
{
  int idx = blockIdx.x * blockDim.x + threadIdx.x;
  if (idx >= Eg * NH) return;
  int e = idx >> 2, hh = idx & 3;
  int d = (e < NE) ? dst[e] : (e - NE);
  float ee = expf(elog[idx] - mfin(mbuf[d * NH + hh]));
  elog[idx] = ee;
  atomicAdd(&sbuf[d * NH + hh], ee);
}

__global__ void gat_edge_agg(const int* __restrict__ src, const int* __restrict__ dst,
                             const float* __restrict__ ee, const float* __restrict__ sbuf,
                             const float* __restrict__ h, float* __restrict__ out, int Eg)
{
  long idx = (long)blockIdx.x * blockDim.x + threadIdx.x;
  if (idx >= (long)Eg * DIM) return;
  int e = (int)(idx >> 7), j = (int)(idx & 127), hh = j >> 5;
  int s = (e < NE) ? src[e] : (e - NE);
  int d = (e < NE) ? dst[e] : (e - NE);
  float alpha = ee[e * NH + hh] / (sbuf[d * NH + hh] + 1e-16f);
  atomicAdd(&out[(size_t)d * DIM + j], alpha * h[(size_t)s * DIM + j]);
}

__global__ void bn_stats(const float* __restrict__ x,
                         float* __restrict__ sum, float* __restrict__ sumsq)
{
  int col = threadIdx.x;
  float s = 0.f, s2 = 0.f;
  for (int r = blockIdx.x; r < N_NODES; r += gridDim.x) {
    float v = x[(size_t)r * DIM + col]; s += v; s2 += v * v;
  }
  atomicAdd(&sum[col], s); atomicAdd(&sumsq[col], s2);
}

__global__ void bn_apply_lrelu(const float* __restrict__ res, const float* __restrict__ x,
                               const float* __restrict__ sum, const float* __restrict__ sumsq,
                               const float* __restrict__ gamma, const float* __restrict__ beta,
                               float* __restrict__ out)
{
  long idx = (long)blockIdx.x * blockDim.x + threadIdx.x;
  if (idx >= (long)N_NODES * DIM) return;
  int j = (int)(idx & 127);
  const float inv_n = 1.f / (float)N_NODES;
  float mu  = sum[j] * inv_n;
  float var = sumsq[j] * inv_n - mu * mu;
  float xn  = (x[idx] - mu) * rsqrtf(var + BN_EPS) * gamma[j] + beta[j];
  out[idx] = lrelu(res[idx] + xn);
}

// a_rel[h,d,f] -> dense head-block-diagonal 128x128 so the per-edge einsum
// becomes a per-node WMMA GEMM
__global__ void pack_blockdiag(const float* __restrict__ rel, float* __restrict__ dmat)
{
  int idx = blockIdx.x * blockDim.x + threadIdx.x;
  if (idx >= DIM * DIM) return;
  int r = idx >> 7, c = idx & 127;
  float v = 0.f;
  if ((r >> 5) == (c >> 5))
    v = rel[(r >> 5) * DH * DH + (r & 31) * DH + (c & 31)];
  dmat[idx] = v;
}

__global__ void hgt_edge_logit(const int* __restrict__ src, const int* __restrict__ dst,
                               const float* __restrict__ kr, const float* __restrict__ q,
                               const float* __restrict__ prel, float scale,
                               float* __restrict__ abuf, float* __restrict__ mbuf)
{
  int idx = blockIdx.x * blockDim.x + threadIdx.x;
  if (idx >= NE * NH) return;
  int e = idx >> 2, hh = idx & 3;
  int s = src[e], d = dst[e];
  const float4* kp = (const float4*)(kr + (size_t)s * DIM + hh * DH);
  const float4* qp = (const float4*)(q  + (size_t)d * DIM + hh * DH);
  float a = 0.f;
  #pragma unroll
  for (int i = 0; i < DH / 4; ++i) {
    float4 kv = kp[i], qv = qp[i];
    a += kv.x * qv.x + kv.y * qv.y + kv.z * qv.z + kv.w * qv.w;
  }
  a *= prel[hh] * scale;
  abuf[idx] = a;
  atomicMaxFloat(&mbuf[d * NH + hh], a);
}

__global__ void hgt_edge_exps(const int* __restrict__ dst,
                              float* __restrict__ abuf, const float* __restrict__ mbuf,
                              float* __restrict__ sbuf)
{
  int idx = blockIdx.x * blockDim.x + threadIdx.x;
  if (idx >= NE * NH) return;
  int e = idx >> 2, hh = idx & 3;
  int d = dst[e];
  float ee = expf(abuf[idx] - mfin(mbuf[d * NH + hh]));
  abuf[idx] = ee;
  atomicAdd(&sbuf[d * NH + hh], ee);
}

__global__ void hgt_edge_agg(const int* __restrict__ src, const int* __restrict__ dst,
                             const float* __restrict__ ee, const float* __restrict__ sbuf,
                             const float* __restrict__ vr, float* __restrict__ out)
{
  long idx = (long)blockIdx.x * blockDim.x + threadIdx.x;
  if (idx >= (long)NE * DIM) return;
  int e = (int)(idx >> 7), j = (int)(idx & 127), hh = j >> 5;
  int s = src[e], d = dst[e];
  float alpha = ee[e * NH + hh] / (sbuf[d * NH + hh] + 1e-16f);
  atomicAdd(&out[(size_t)d * DIM + j], alpha * vr[(size_t)s * DIM + j]);
}

__global__ void hgt_combine(const float* __restrict__ hout, const float* __restrict__ x2,
                            const float* __restrict__ skip, float* __restrict__ out)
{
  long idx = (long)blockIdx.x * blockDim.x + threadIdx.x;
  if (idx >= (long)N_NODES * DIM) return;
  float p = 1.f / (1.f + expf(-skip[0]));
  out[idx] = p * hout[idx] + (1.f - p) * x2[idx];
}

__global__ void film_final(const float* __restrict__ x3, const float* __restrict__ inj,
                           const float* __restrict__ h2, const int* __restrict__ batch,
                           float* __restrict__ out)
{
  long idx = (long)blockIdx.x * blockDim.x + threadIdx.x;
  if (idx >= (long)N_NODES * DIM) return;
  int n = (int)(idx >> 7), j = (int)(idx & 127);
  int b = batch[n];
  float g  = h2[(size_t)b * 2 * DIM + j];
  float be = h2[(size_t)b * 2 * DIM + DIM + j];
  out[idx] = g * (x3[idx] + inj[idx]) + be;
}

// ---------------------------------------------------------------------------
extern "C" void kernel_launch(void* const* d_in, const int* in_sizes, int n_in,
                              void* d_out, int out_size, void* d_ws, size_t ws_size,
                              hipStream_t stream)
{
  (void)in_sizes; (void)n_in; (void)out_size; (void)ws_size;
  const float* x_cell = (const float*)d_in[0];
  const float* x_emb  = (const float*)d_in[1];
  const float* z_loc  = (const float*)d_in[2];
  const float* g1W    = (const float*)d_in[3];
  const float* g1as   = (const float*)d_in[4];
  const float* g1ad   = (const float*)d_in[5];
  const float* g2W    = (const float*)d_in[7];
  const float* g2as   = (const float*)d_in[8];
  const float* g2ad   = (const float*)d_in[9];
  const float* bn1g   = (const float*)d_in[11];
  const float* bn1b   = (const float*)d_in[12];
  const float* bn2g   = (const float*)d_in[13];
  const float* bn2b   = (const float*)d_in[14];
  const float* bnhg   = (const float*)d_in[15];
  const float* bnhb   = (const float*)d_in[16];
  const float* hWk    = (const float*)d_in[17];
  const float* hbk    = (const float*)d_in[18];
  const float* hWq    = (const float*)d_in[19];
  const float* hbq    = (const float*)d_in[20];
  const float* hWv    = (const float*)d_in[21];
  const float* hbv    = (const float*)d_in[22];
  const float* hWa    = (const float*)d_in[23];
  const float* hba    = (const float*)d_in[24];
  const float* hskip  = (const float*)d_in[25];
  const float* arel   = (const float*)d_in[26];
  const float* mrel   = (const float*)d_in[27];
  const float* prel   = (const float*)d_in[28];
  const float* injW   = (const float*)d_in[29];
  const float* injb   = (const float*)d_in[30];
  const float* fW1    = (const float*)d_in[31];
  const float* fb1    = (const float*)d_in[32];
  const float* fW2    = (const float*)d_in[33];
  const float* fb2    = (const float*)d_in[34];
  const int*   ei0    = (const int*)d_in[35];   // [2][NE]
  const int*   ei1    = (const int*)d_in[36];
  const int*   batch  = (const int*)d_in[37];

  const size_t ND = (size_t)N_NODES * DIM;
  float* ws = (float*)d_ws;
  float* hbuf  = ws; ws += ND;
  float* gsum  = ws; ws += ND;
  float* x1    = ws; ws += ND;
  float* x2    = ws; ws += ND;
  float* x3    = ws; ws += ND;
  float* kbuf  = ws; ws += ND;
  float* qbuf  = ws; ws += ND;
  float* vbuf  = ws; ws += ND;
  float* krbuf = ws; ws += ND;                      // kr, later vr
  float* alsrc = ws; ws += (size_t)N_NODES * NH;
  float* aldst = ws; ws += (size_t)N_NODES * NH;
  float* mbuf  = ws; ws += (size_t)N_NODES * NH;
  float* sbuf  = ws; ws += (size_t)N_NODES * NH;
  float* elog  = ws; ws += (size_t)2 * NE * NH;     // >= (NE+N)*NH too
  float* dmat  = ws; ws += (size_t)DIM * DIM;
  float* csum  = ws; ws += DIM;
  float* csq   = ws; ws += DIM;
  float* f1    = ws; ws += (size_t)NB * 2 * DIM;
  float* f2    = ws; ws += (size_t)NB * 2 * DIM;

  const dim3 gN(N_NODES / TILE_M, 1);
  const int  ntNH  = N_NODES * NH;
  const int  Eg    = NE + N_NODES;
  const long ndTot = (long)ND;

  auto run_gat = [&](const float* xin, const float* Wall,
                     const float* asall, const float* adall,
                     const float* bng, const float* bnb, float* xout) {
    hipMemsetAsync(gsum, 0, ND * sizeof(float), stream);
    for (int t = 0; t < 2; ++t) {
      gemm_wmma<128, 128><<<gN, 256, 0, stream>>>(
          xin, Wall + (size_t)t * DIM * DIM, nullptr, hbuf);
      gat_logits<<<(ntNH + 255) / 256, 256, 0, stream>>>(
          hbuf, asall + t * NH * DH, adall + t * NH * DH, alsrc, aldst);
      fill_f32<<<(ntNH + 255) / 256, 256, 0, stream>>>(mbuf, -INFINITY, ntNH);
      hipMemsetAsync(sbuf, 0, (size_t)ntNH * sizeof(float), stream);
      const int* src = (t == 0 ? ei0 : ei1);
      const int* dst = src + NE;
      long ne4 = (long)Eg * NH;
      gat_edge_max<<<(int)((ne4 + 255) / 256), 256, 0, stream>>>(
          src, dst, alsrc, aldst, elog, mbuf, Eg);
      gat_edge_exps<<<(int)((ne4 + 255) / 256), 256, 0, stream>>>(
          dst, elog, mbuf, sbuf, Eg);
      long nagg = (long)Eg * DIM;
      gat_edge_agg<<<(int)((nagg + 255) / 256), 256, 0, stream>>>(
          src, dst, elog, sbuf, hbuf, gsum, Eg);
    }
    // gat bias is a constant column shift -> cancels exactly inside BN.
    hipMemsetAsync(csum, 0, DIM * sizeof(float), stream);
    hipMemsetAsync(csq,  0, DIM * sizeof(float), stream);
    bn_stats<<<512, 128, 0, stream>>>(gsum, csum, csq);
    bn_apply_lrelu<<<(int)((ndTot + 255) / 256), 256, 0, stream>>>(
        xin, gsum, csum, csq, bng, bnb, xout);
  };

  run_gat(x_cell, g1W, g1as, g1ad, bn1g, bn1b, x1);
  run_gat(x1,     g2W, g2as, g2ad, bn2g, bn2b, x2);

  // ---- HGT ----
  gemm_wmma<128, 128><<<gN, 256, 0, stream>>>(x2, hWk, hbk, kbuf);
  gemm_wmma<128, 128><<<gN, 256, 0, stream>>>(x2, hWq, hbq, qbuf);
  gemm_wmma<128, 128><<<gN, 256, 0, stream>>>(x2, hWv, hbv, vbuf);

  fill_f32<<<(ntNH + 255) / 256, 256, 0, stream>>>(mbuf, -INFINITY, ntNH);
  hipMemsetAsync(sbuf, 0, (size_t)ntNH * sizeof(float), stream);
  const float scale = 0.17677669529663687f;   // 1/sqrt(32)
  const long nE4 = (long)NE * NH;

  for (int t = 0; t < 2; ++t) {               // logits + segment max
    pack_blockdiag<<<(DIM * DIM) / 256, 256, 0, stream>>>(
        arel + (size_t)t * NH * DH * DH, dmat);
    gemm_wmma<128, 128><<<gN, 256, 0, stream>>>(kbuf, dmat, nullptr, krbuf);
    const int* src = (t == 0 ? ei0 : ei1);
    const int* dst = src + NE;
    hgt_edge_logit<<<(int)((nE4 + 255) / 256), 256, 0, stream>>>(
        src, dst, krbuf, qbuf, prel + t * NH, scale,
        elog + (size_t)t * NE * NH, mbuf);
  }
  for (int t = 0; t < 2; ++t) {               // exp + segment sum (joint softmax)
    const int* dst = (t == 0 ? ei0 : ei1) + NE;
    hgt_edge_exps<<<(int)((nE4 + 255) / 256), 256, 0, stream>>>(
        dst, elog + (size_t)t * NE * NH, mbuf, sbuf);
  }
  hipMemsetAsync(gsum, 0, ND * sizeof(float), stream);
  for (int t = 0; t < 2; ++t) {               // v_rel + weighted scatter
    pack_blockdiag<<<(DIM * DIM) / 256, 256, 0, stream>>>(
        mrel + (size_t)t * NH * DH * DH, dmat);
    gemm_wmma<128, 128><<<gN, 256, 0, stream>>>(vbuf, dmat, nullptr, krbuf);
    const int* src = (t == 0 ? ei0 : ei1);
    const int* dst = src + NE;
    long nagg = (long)NE * DIM;
    hgt_edge_agg<<<(int)((nagg + 255) / 256), 256, 0, stream>>>(
        src, dst, elog + (size_t)t * NE * NH, sbuf, krbuf, gsum);
  }
  gelu_k<<<(int)((ndTot + 255) / 256), 256, 0, stream>>>(gsum, qbuf, ndTot);
  gemm_wmma<128, 128><<<gN, 256, 0, stream>>>(qbuf, hWa, hba, hbuf);
  hgt_combine<<<(int)((ndTot + 255) / 256), 256, 0, stream>>>(hbuf, x2, hskip, kbuf);

  hipMemsetAsync(csum, 0, DIM * sizeof(float), stream);
  hipMemsetAsync(csq,  0, DIM * sizeof(float), stream);
  bn_stats<<<512, 128, 0, stream>>>(kbuf, csum, csq);
  bn_apply_lrelu<<<(int)((ndTot + 255) / 256), 256, 0, stream>>>(
      x2, kbuf, csum, csq, bnhg, bnhb, x3);

  // injection + FiLM head
  gemm_wmma<128, 128><<<gN, 256, 0, stream>>>(x_emb, injW, injb, vbuf);
  gemm_wmma<128, 256><<<dim3(NB / TILE_M, 2), 256, 0, stream>>>(z_loc, fW1, fb1, f1);
  gelu_k<<<(NB * 2 * DIM + 255) / 256, 256, 0, stream>>>(f1, f1, NB * 2 * DIM);
  gemm_wmma<256, 256><<<dim3(NB / TILE_M, 2), 256, 0, stream>>>(f1, fW2, fb2, f2);

  film_final<<<(int)((ndTot + 255) / 256), 256, 0, stream>>>(
      x3, vbuf, f2, batch, (float*)d_out);
}